// DINO_66494683677070
// MI455X (gfx1250) — compile-verified
//
#include <hip/hip_runtime.h>
#include <hip/hip_bf16.h>
#include <math.h>

// ---------------------------------------------------------------------------
// CDNA5 (gfx1250) implementation of the GP marginal-likelihood pipeline.
// All O(n^3) work (GEMM / SYRK / TRSM-update) goes through a single WMMA-f32
// tile kernel using V_WMMA_F32_16X16X4_F32 (wave32).  Each wave computes a
// 64x64 C super-tile (4x4 WMMA tiles): per K-step of 4 it loads 4 A-frags +
// 4 B-frags (16 dwords/lane) and issues 16 WMMAs -> 16 FLOP/byte from cache,
// half a fragment load per WMMA.
// ---------------------------------------------------------------------------

#if defined(__has_builtin)
#  if __has_builtin(__builtin_amdgcn_wmma_f32_16x16x4_f32)
#    define HAVE_WMMA_F32 1
#  else
#    define HAVE_WMMA_F32 0
#  endif
#else
#  define HAVE_WMMA_F32 0
#endif

typedef float v2f __attribute__((ext_vector_type(2)));
typedef float v8f __attribute__((ext_vector_type(8)));

#define NSRC 2048
#define NTGT 2048
#define NBASE 512
#define DIMX 256
#define NFEAT 8
#define CHB 64   // Cholesky / TRSM block size

// ---------------------------------------------------------------------------
// WMMA f32 GEMM tile kernel: wave computes a 64x64 C super-tile (4x4 WMMAs).
//   C(MxN) = epilogue( op(A)(MxK) @ op(B)(KxN) )
// TA: A_eff[m][k] = A[k*lda+m]   (else A[m*lda+k])
// TB: B_eff[k][n] = B[n*ldb+k]   (else B[k*ldb+n])
// EPI 0: C = AB
// EPI 1: C = E .* (AB) + clamp(noise)*I   (noise nullable -> no diagonal)
// EPI 2: C = C0 - AB                      (C0 may alias C)
// M,N multiples of 64; K multiple of 4.
// Fragment layouts per CDNA5 ISA 7.12.2:
//   A 16x4: lane(l&15)=M, VGPR v holds K = 2*(l>>4)+v
//   B 4x16: lane(l&15)=N, VGPR v holds K = 2*(l>>4)+v
//   C/D:    lane(l&15)=N, VGPR v holds M = v + 8*(l>>4)
// ---------------------------------------------------------------------------
template <bool TA, bool TB, int EPI>
__global__ __launch_bounds__(256) void gemm_wmma_kernel(
    const float* __restrict__ A, int lda,
    const float* __restrict__ B, int ldb,
    const float* __restrict__ E,
    const float* __restrict__ noise,
    const float* C0,
    float* C, int ldc,
    int M, int N, int K)
{
  const int lane = threadIdx.x & 31;
  const int wid  = threadIdx.x >> 5;
  const int tilesN = N >> 6;                       // 64-wide super-tiles
  const long totalTiles = (long)(M >> 6) * (long)tilesN;
  const long tile = (long)blockIdx.x * 8 + wid;
  if (tile >= totalTiles) return;                  // wave-uniform guard

  const int m0 = (int)(tile / tilesN) << 6;
  const int n0 = (int)(tile % tilesN) << 6;
  const int li = lane & 15;
  const int lg = lane >> 4;

  v8f acc[4][4];
  #pragma unroll
  for (int tm = 0; tm < 4; ++tm)
    #pragma unroll
    for (int tn = 0; tn < 4; ++tn)
      acc[tm][tn] = (v8f){0.f,0.f,0.f,0.f,0.f,0.f,0.f,0.f};

  for (int k0 = 0; k0 < K; k0 += 4) {
#if HAVE_WMMA_F32
    v2f av[4], bv[4];
    {
      const int am = m0 + li;
      const int ak = k0 + (lg << 1);
      #pragma unroll
      for (int t = 0; t < 4; ++t) {
        if (!TA) {
          const float* p = A + (size_t)(am + (t << 4)) * lda + ak;
          av[t][0] = p[0]; av[t][1] = p[1];
        } else {
          const float* p = A + (size_t)ak * lda + am + (t << 4);
          av[t][0] = p[0]; av[t][1] = p[lda];
        }
      }
      const int bn = n0 + li;
      const int bk = k0 + (lg << 1);
      #pragma unroll
      for (int t = 0; t < 4; ++t) {
        if (!TB) {
          const float* p = B + (size_t)bk * ldb + bn + (t << 4);
          bv[t][0] = p[0]; bv[t][1] = p[ldb];
        } else {
          const float* p = B + (size_t)(bn + (t << 4)) * ldb + bk;
          bv[t][0] = p[0]; bv[t][1] = p[1];
        }
      }
    }
    #pragma unroll
    for (int tm = 0; tm < 4; ++tm)
      #pragma unroll
      for (int tn = 0; tn < 4; ++tn)
        acc[tm][tn] = __builtin_amdgcn_wmma_f32_16x16x4_f32(
            false, av[tm], false, bv[tn], (short)0, acc[tm][tn], false, false);
#else
    // per-lane FMA fallback matching the C fragment layout
    #pragma unroll
    for (int kk = 0; kk < 4; ++kk) {
      const int k = k0 + kk;
      float bvv[4], avv[4];
      #pragma unroll
      for (int t = 0; t < 4; ++t) {
        const int n = n0 + li + (t << 4);
        bvv[t] = TB ? B[(size_t)n * ldb + k] : B[(size_t)k * ldb + n];
      }
      #pragma unroll
      for (int v = 0; v < 8; ++v) {
        #pragma unroll
        for (int t = 0; t < 4; ++t) {
          const int m = m0 + (t << 4) + v + (lg << 3);
          avv[t] = TA ? A[(size_t)k * lda + m] : A[(size_t)m * lda + k];
        }
        #pragma unroll
        for (int tm = 0; tm < 4; ++tm)
          #pragma unroll
          for (int tn = 0; tn < 4; ++tn)
            acc[tm][tn][v] += avv[tm] * bvv[tn];
      }
    }
#endif
  }

  // epilogue over the 4x4 WMMA tiles
  #pragma unroll
  for (int tm = 0; tm < 4; ++tm) {
    #pragma unroll
    for (int tn = 0; tn < 4; ++tn) {
      #pragma unroll
      for (int v = 0; v < 8; ++v) {
        const int m = m0 + (tm << 4) + v + (lg << 3);
        const int n = n0 + (tn << 4) + li;
        const size_t idx = (size_t)m * ldc + n;
        float r = acc[tm][tn][v];
        if (EPI == 1) {
          r *= E[idx];
          if (noise != nullptr && m == n)
            r += fminf(fmaxf(noise[0], 1e-5f), 1.0f);
          C[idx] = r;
        } else if (EPI == 2) {
          C[idx] = C0[idx] - r;
        } else {
          C[idx] = r;
        }
      }
    }
  }
}

// ---------------------------------------------------------------------------
// feat_dot: out[row] = sum_ft (x[row,:] @ Wf[ft,:] + bf[ft]) * wv[ft]
// One 256-thread block per row; DIM == 256.
// ---------------------------------------------------------------------------
__global__ __launch_bounds__(256) void feat_dot_kernel(
    const float* __restrict__ x, const float* __restrict__ Wf,
    const float* __restrict__ bf, const float* __restrict__ wv,
    float* __restrict__ out)
{
  __shared__ float red[256];
  __shared__ float fsum[NFEAT];
  const int row = blockIdx.x;
  const int tid = threadIdx.x;
  const float xv = x[(size_t)row * DIMX + tid];
  float acc[NFEAT];
  #pragma unroll
  for (int ft = 0; ft < NFEAT; ++ft) acc[ft] = xv * Wf[ft * DIMX + tid];
  for (int ft = 0; ft < NFEAT; ++ft) {
    red[tid] = acc[ft];
    __syncthreads();
    for (int s = 128; s > 0; s >>= 1) {
      if (tid < s) red[tid] += red[tid + s];
      __syncthreads();
    }
    if (tid == 0) fsum[ft] = red[0] + bf[ft];
    __syncthreads();
  }
  if (tid == 0) {
    float d = 0.f;
    #pragma unroll
    for (int ft = 0; ft < NFEAT; ++ft) d += fsum[ft] * wv[ft];
    out[row] = d;
  }
}

// ---------------------------------------------------------------------------
// weights: W[i,j] = elu(fdot[i]+gdot[j]+bw) ; then row-normalize by mean.
// ---------------------------------------------------------------------------
__global__ __launch_bounds__(256) void weights_kernel(
    const float* __restrict__ fdot, const float* __restrict__ gdot,
    const float* __restrict__ bw, float* __restrict__ W)
{
  __shared__ float wbuf[NBASE];
  __shared__ float red[256];
  const int i = blockIdx.x;
  const int tid = threadIdx.x;
  const float fv = fdot[i];
  const float b = bw[0];
  float p = 0.f;
  for (int j = tid; j < NBASE; j += 256) {
    const float raw = fv + gdot[j] + b;
    const float w = raw > 0.f ? raw : expm1f(raw);
    wbuf[j] = w;
    p += w;
  }
  red[tid] = p;
  __syncthreads();
  for (int s = 128; s > 0; s >>= 1) {
    if (tid < s) red[tid] += red[tid + s];
    __syncthreads();
  }
  const float scale = (float)NBASE / red[0];
  for (int j = tid; j < NBASE; j += 256)
    W[(size_t)i * NBASE + j] = wbuf[j] * scale;
}

// ---------------------------------------------------------------------------
// POTF2: in-place Cholesky of a CHB x CHB diagonal block (one workgroup).
// ---------------------------------------------------------------------------
__global__ __launch_bounds__(256) void potf2_kernel(float* A, int lda)
{
  __shared__ float t[CHB][CHB + 1];
  const int tid = threadIdx.x;
  for (int idx = tid; idx < CHB * CHB; idx += 256)
    t[idx / CHB][idx % CHB] = A[(size_t)(idx / CHB) * lda + (idx % CHB)];
  __syncthreads();
  for (int j = 0; j < CHB; ++j) {
    if (tid == 0) {
      float s = t[j][j];
      for (int q = 0; q < j; ++q) s -= t[j][q] * t[j][q];
      t[j][j] = sqrtf(fmaxf(s, 1e-20f));
    }
    __syncthreads();
    const float inv = 1.0f / t[j][j];
    for (int i = j + 1 + tid; i < CHB; i += 256) {
      float s = t[i][j];
      for (int q = 0; q < j; ++q) s -= t[i][q] * t[j][q];
      t[i][j] = s * inv;
    }
    __syncthreads();
  }
  for (int idx = tid; idx < CHB * CHB; idx += 256)
    A[(size_t)(idx / CHB) * lda + (idx % CHB)] = t[idx / CHB][idx % CHB];
}

// ---------------------------------------------------------------------------
// TRSM (right): A21 <- A21 * L11^{-T}.  One thread per row, row in registers.
// ---------------------------------------------------------------------------
__global__ __launch_bounds__(128) void trsm_right_kernel(
    float* A21, const float* __restrict__ L11, int lda, int nrows)
{
  __shared__ float L[CHB][CHB + 1];
  const int tid = threadIdx.x;
  for (int idx = tid; idx < CHB * CHB; idx += 128)
    L[idx / CHB][idx % CHB] = L11[(size_t)(idx / CHB) * lda + (idx % CHB)];
  __syncthreads();
  const int row = blockIdx.x * 128 + tid;
  if (row >= nrows) return;
  float x[CHB];
  #pragma unroll
  for (int j = 0; j < CHB; ++j) x[j] = A21[(size_t)row * lda + j];
  #pragma unroll
  for (int j = 0; j < CHB; ++j) {
    float s = x[j];
    #pragma unroll
    for (int i = 0; i < j; ++i) s -= x[i] * L[j][i];
    x[j] = s / L[j][j];
  }
  #pragma unroll
  for (int j = 0; j < CHB; ++j) A21[(size_t)row * lda + j] = x[j];
}

// ---------------------------------------------------------------------------
// TRSM (left) on a 64-row strip: B <- L11^{-1} B.  One thread per column.
// ---------------------------------------------------------------------------
__global__ __launch_bounds__(256) void trsm_left_kernel(
    const float* __restrict__ L11, int ldl, float* B, int ldb, int ncols)
{
  __shared__ float L[CHB][CHB + 1];
  const int tid = threadIdx.x;
  for (int idx = tid; idx < CHB * CHB; idx += 256)
    L[idx / CHB][idx % CHB] = L11[(size_t)(idx / CHB) * ldl + (idx % CHB)];
  __syncthreads();
  const int col = blockIdx.x * 256 + tid;
  if (col >= ncols) return;
  float x[CHB];
  #pragma unroll
  for (int r = 0; r < CHB; ++r) x[r] = B[(size_t)r * ldb + col];
  #pragma unroll
  for (int r = 0; r < CHB; ++r) {
    float s = x[r];
    #pragma unroll
    for (int q = 0; q < r; ++q) s -= L[r][q] * x[q];
    x[r] = s / L[r][r];
  }
  #pragma unroll
  for (int r = 0; r < CHB; ++r) B[(size_t)r * ldb + col] = x[r];
}

// ---------------------------------------------------------------------------
// Dense forward substitution for a single RHS vector (one workgroup).
// y = L^{-1} b ; optionally quad[0] = sum(y^2).
// ---------------------------------------------------------------------------
__global__ __launch_bounds__(256) void solve_vec_kernel(
    const float* __restrict__ L, int n, int ldl,
    const float* __restrict__ b, float* __restrict__ y, float* quad)
{
  __shared__ float yv[NSRC];
  __shared__ float red[256];
  const int tid = threadIdx.x;
  for (int i = tid; i < n; i += 256) yv[i] = b[i];
  __syncthreads();
  for (int i = 0; i < n; ++i) {
    float p = 0.f;
    for (int j = tid; j < i; j += 256) p += L[(size_t)i * ldl + j] * yv[j];
    red[tid] = p;
    __syncthreads();
    for (int s = 128; s > 0; s >>= 1) {
      if (tid < s) red[tid] += red[tid + s];
      __syncthreads();
    }
    if (tid == 0) yv[i] = (yv[i] - red[0]) / L[(size_t)i * ldl + i];
    __syncthreads();
  }
  for (int i = tid; i < n; i += 256) y[i] = yv[i];
  if (quad != nullptr) {
    float p = 0.f;
    for (int i = tid; i < n; i += 256) p += yv[i] * yv[i];
    red[tid] = p;
    __syncthreads();
    for (int s = 128; s > 0; s >>= 1) {
      if (tid < s) red[tid] += red[tid + s];
      __syncthreads();
    }
    if (tid == 0) quad[0] = red[0];
  }
}

// r[n] = ty[n] - sum_k X[k][n] * alpha[k]      (mu = X^T alpha)
__global__ __launch_bounds__(256) void mu_r_kernel(
    const float* __restrict__ X, int ldx, const float* __restrict__ alpha,
    const float* __restrict__ ty, float* __restrict__ r, int n, int krows)
{
  const int nIdx = blockIdx.x * 256 + threadIdx.x;
  if (nIdx >= n) return;
  float acc = 0.f;
  for (int k = 0; k < krows; ++k)
    acc += X[(size_t)k * ldx + nIdx] * alpha[k];
  r[nIdx] = ty[nIdx] - acc;
}

// loss = 0.5*sum(log(diag(Lcov))) + 0.5*quad + 0.5*NS*log(2*pi)
__global__ __launch_bounds__(256) void loss_kernel(
    const float* __restrict__ Lcov, int n, int ldl,
    const float* __restrict__ quad, float* __restrict__ out)
{
  __shared__ float red[256];
  const int tid = threadIdx.x;
  float p = 0.f;
  for (int i = tid; i < n; i += 256)
    p += logf(fmaxf(Lcov[(size_t)i * ldl + i], 1e-30f));
  red[tid] = p;
  __syncthreads();
  for (int s = 128; s > 0; s >>= 1) {
    if (tid < s) red[tid] += red[tid + s];
    __syncthreads();
  }
  if (tid == 0)
    out[0] = 0.5f * red[0] + 0.5f * quad[0]
           + 0.5f * (float)NSRC * 1.8378770664093453f; // log(2*pi)
}

// ---------------------------------------------------------------------------
// Host side
// ---------------------------------------------------------------------------
static inline unsigned gemm_blocks(int M, int N) {
  long tiles = (long)(M / 64) * (long)(N / 64);   // 64x64 super-tiles
  return (unsigned)((tiles + 7) / 8);
}

extern "C" void kernel_launch(void* const* d_in, const int* in_sizes, int n_in,
                              void* d_out, int out_size, void* d_ws, size_t ws_size,
                              hipStream_t stream) {
  (void)in_sizes; (void)n_in; (void)out_size; (void)ws_size;

  const float* source_x = (const float*)d_in[0];
  const float* source_y = (const float*)d_in[1];
  const float* target_x = (const float*)d_in[2];
  const float* target_y = (const float*)d_in[3];
  const float* k_ss     = (const float*)d_in[4];
  const float* k_tt     = (const float*)d_in[5];
  const float* k_st     = (const float*)d_in[6];
  const float* noise_s  = (const float*)d_in[7];
  const float* noise_t  = (const float*)d_in[8];
  const float* Wf       = (const float*)d_in[9];
  const float* bf       = (const float*)d_in[10];
  const float* Ws       = (const float*)d_in[11];
  const float* bs       = (const float*)d_in[12];
  const float* Wt       = (const float*)d_in[13];
  const float* bt       = (const float*)d_in[14];
  const float* Kb       = (const float*)d_in[15];
  const float* base_s   = (const float*)d_in[16];
  const float* base_t   = (const float*)d_in[17];

  float* W = (float*)d_ws;
  const size_t SZW = (size_t)NSRC * NBASE;           // 2048*512
  const size_t SZK = (size_t)NSRC * NSRC;            // 2048*2048
  float* ws_w  = W;                                  // 2048x512
  float* wt_w  = ws_w + SZW;
  float* wsK   = wt_w + SZW;
  float* wtK   = wsK + SZW;
  float* Ksrc  = wtK + SZW;                          // -> L_s after chol
  float* Ktgt  = Ksrc + SZK;                         // -> cov -> L_cov
  float* Kcos  = Ktgt + SZK;                         // -> X = L_s^{-1} K_cos
  float* fdot_s = Kcos + SZK;
  float* fdot_t = fdot_s + NSRC;
  float* gdot_s = fdot_t + NTGT;
  float* gdot_t = gdot_s + NBASE;
  float* alpha  = gdot_t + NBASE;
  float* rvec   = alpha + NSRC;
  float* zvec   = rvec + NTGT;
  float* quad   = zvec + NTGT;

  // ---- feature maps and kernel weights ----
  feat_dot_kernel<<<NSRC, 256, 0, stream>>>(source_x, Wf, bf, Ws,         fdot_s);
  feat_dot_kernel<<<NBASE, 256, 0, stream>>>(base_s,  Wf, bf, Ws + NFEAT, gdot_s);
  feat_dot_kernel<<<NTGT, 256, 0, stream>>>(target_x, Wf, bf, Wt,         fdot_t);
  feat_dot_kernel<<<NBASE, 256, 0, stream>>>(base_t,  Wf, bf, Wt + NFEAT, gdot_t);
  weights_kernel<<<NSRC, 256, 0, stream>>>(fdot_s, gdot_s, bs, ws_w);
  weights_kernel<<<NTGT, 256, 0, stream>>>(fdot_t, gdot_t, bt, wt_w);

  // ---- wsK = ws @ Kb ; wtK = wt @ Kb ----
  gemm_wmma_kernel<false,false,0><<<gemm_blocks(NSRC, NBASE), 256, 0, stream>>>(
      ws_w, NBASE, Kb, NBASE, nullptr, nullptr, nullptr, wsK, NBASE,
      NSRC, NBASE, NBASE);
  gemm_wmma_kernel<false,false,0><<<gemm_blocks(NTGT, NBASE), 256, 0, stream>>>(
      wt_w, NBASE, Kb, NBASE, nullptr, nullptr, nullptr, wtK, NBASE,
      NTGT, NBASE, NBASE);

  // ---- K_src, K_tgt, K_cos ----
  gemm_wmma_kernel<false,true,1><<<gemm_blocks(NSRC, NSRC), 256, 0, stream>>>(
      wsK, NBASE, ws_w, NBASE, k_ss, noise_s, nullptr, Ksrc, NSRC,
      NSRC, NSRC, NBASE);
  gemm_wmma_kernel<false,true,1><<<gemm_blocks(NTGT, NTGT), 256, 0, stream>>>(
      wtK, NBASE, wt_w, NBASE, k_tt, noise_t, nullptr, Ktgt, NTGT,
      NTGT, NTGT, NBASE);
  gemm_wmma_kernel<false,true,1><<<gemm_blocks(NSRC, NTGT), 256, 0, stream>>>(
      wsK, NBASE, wt_w, NBASE, k_st, nullptr, nullptr, Kcos, NTGT,
      NSRC, NTGT, NBASE);

  // ---- blocked Cholesky (in-place, lower) ----
  auto cholesky = [&](float* Amat, int n) {
    for (int k = 0; k < n; k += CHB) {
      potf2_kernel<<<1, 256, 0, stream>>>(Amat + (size_t)k * n + k, n);
      const int trail = n - k - CHB;
      if (trail > 0) {
        float* A21 = Amat + (size_t)(k + CHB) * n + k;
        trsm_right_kernel<<<(trail + 127) / 128, 128, 0, stream>>>(
            A21, Amat + (size_t)k * n + k, n, trail);
        float* Ctr = Amat + (size_t)(k + CHB) * n + (k + CHB);
        gemm_wmma_kernel<false,true,2><<<gemm_blocks(trail, trail), 256, 0, stream>>>(
            A21, n, A21, n, nullptr, nullptr, Ctr, Ctr, n,
            trail, trail, CHB);
      }
    }
  };

  cholesky(Ksrc, NSRC);   // Ksrc now holds L_s (lower)

  // ---- X = L_s^{-1} K_cos  (blocked forward TRSM, in-place in Kcos) ----
  for (int i = 0; i < NSRC; i += CHB) {
    if (i > 0) {
      // X[i:i+64,:] -= L_s[i:i+64, 0:i] @ X[0:i, :]
      gemm_wmma_kernel<false,false,2><<<gemm_blocks(CHB, NTGT), 256, 0, stream>>>(
          Ksrc + (size_t)i * NSRC, NSRC, Kcos, NTGT, nullptr, nullptr,
          Kcos + (size_t)i * NTGT, Kcos + (size_t)i * NTGT, NTGT,
          CHB, NTGT, i);
    }
    trsm_left_kernel<<<(NTGT + 255) / 256, 256, 0, stream>>>(
        Ksrc + (size_t)i * NSRC + i, NSRC,
        Kcos + (size_t)i * NTGT, NTGT, NTGT);
  }

  // ---- cov = K_tgt - X^T X  (in-place in Ktgt) ----
  gemm_wmma_kernel<true,false,2><<<gemm_blocks(NTGT, NTGT), 256, 0, stream>>>(
      Kcos, NTGT, Kcos, NTGT, nullptr, nullptr, Ktgt, Ktgt, NTGT,
      NTGT, NTGT, NSRC);

  cholesky(Ktgt, NTGT);   // Ktgt now holds L_cov (lower)

  // ---- alpha = L_s^{-1} source_y ; r = target_y - X^T alpha ----
  solve_vec_kernel<<<1, 256, 0, stream>>>(Ksrc, NSRC, NSRC, source_y, alpha, nullptr);
  mu_r_kernel<<<(NTGT + 255) / 256, 256, 0, stream>>>(
      Kcos, NTGT, alpha, target_y, rvec, NTGT, NSRC);

  // ---- z = L_cov^{-1} r ; quad = |z|^2 ----
  solve_vec_kernel<<<1, 256, 0, stream>>>(Ktgt, NTGT, NTGT, rvec, zvec, quad);

  // ---- loss ----
  loss_kernel<<<1, 256, 0, stream>>>(Ktgt, NTGT, NTGT, quad, (float*)d_out);
}